// BoundingBoxRegression_62577673503352
// MI455X (gfx1250) — compile-verified
//
#include <hip/hip_runtime.h>
#include <math.h>

#define B_ 8
#define C_ 16
#define H_ 384
#define W_ 384
#define HW_ (H_*W_)
#define KS 11
#define PADK 5

#define TH 12
#define TW 32
#define TROWS (TH + 10)            // 22
#define TCOLS (TW + 10)            // 42
#define TILE_PER_C (TROWS*TCOLS)   // 924
#define NTILE (C_*TILE_PER_C)      // 14784 floats
#define NTHREADS (TW*TH)           // 384 threads = 12 wave32
#define OUT1_ELEMS (B_*3*C_*HW_)   // 56623104

// fast sigmoid: v_exp_f32 + v_rcp_f32 (~4 VALU). Used identically in the max
// pass and the main pass so the (s > max-0.005) comparison is self-consistent.
__device__ __forceinline__ float sigmoidf_(float x) {
  float e = __expf(-x);                      // v_mul + v_exp
  return __builtin_amdgcn_rcpf(1.0f + e);    // v_add + v_rcp
}

__device__ __forceinline__ float waveSum(float v) {
  v += __shfl_xor(v, 16, 32);
  v += __shfl_xor(v,  8, 32);
  v += __shfl_xor(v,  4, 32);
  v += __shfl_xor(v,  2, 32);
  v += __shfl_xor(v,  1, 32);
  return v;
}

// ---------------- K0: zero workspace accumulators ----------------
__global__ void k_init(float* ws) {
  int i = blockIdx.x * blockDim.x + threadIdx.x;
  if (i < 128 + 128*6) ws[i] = 0.0f;
}

// ---------------- K1: per-(b,c) max of sigmoid(x) ----------------
__global__ __launch_bounds__(256) void k_max(const float* __restrict__ x,
                                             unsigned* __restrict__ wsMbits) {
  const int bc    = blockIdx.y;        // 0..127
  const int split = blockIdx.x;        // 0..7
  const int tid   = threadIdx.x;
  const int chunk = HW_ / 8;           // 18432
  const float* p  = x + (size_t)bc * HW_ + (size_t)split * chunk;
  float m = 0.0f;
  for (int i = tid; i < chunk; i += 256)
    m = fmaxf(m, sigmoidf_(p[i]));
  for (int off = 16; off > 0; off >>= 1)
    m = fmaxf(m, __shfl_xor(m, off, 32));
  __shared__ float red[8];
  if ((tid & 31) == 0) red[tid >> 5] = m;
  __syncthreads();
  if (tid == 0) {
    float mm = red[0];
    for (int i = 1; i < 8; i++) mm = fmaxf(mm, red[i]);
    // sigmoid > 0: positive-float bit patterns are order-isomorphic to uints
    atomicMax(&wsMbits[bc], __float_as_uint(mm));
  }
}

// ---------------- K2: fused main pass ----------------
__global__ __launch_bounds__(NTHREADS) void k_main(
    const float* __restrict__ x,
    const float* __restrict__ wk, const float* __restrict__ wbp,
    const float* __restrict__ hk, const float* __restrict__ hbp,
    const float* __restrict__ skp, const float* __restrict__ sbp,
    const float* __restrict__ wsM,   // 128 plane maxima (bit-punned floats)
    float* __restrict__ gacc,        // 128*6 global accumulators
    float* __restrict__ out1)        // (8,48,384,384)
{
  __shared__ float smem[NTILE + C_*6];
  float* tiles = smem;
  float* accum = smem + NTILE;

  const int tx = threadIdx.x, ty = threadIdx.y;
  const int tid = ty * TW + tx;
  const int w0 = blockIdx.x * TW;
  const int h0 = blockIdx.y * TH;
  const int b  = blockIdx.z;

  if (tid < C_*6) accum[tid] = 0.0f;

  // ---- async-stage all 16 channel halo tiles: global -> LDS, no VGPR round-trip ----
  // low 32 bits of the flat address of a shared object == LDS byte offset
  const unsigned ldsbase = (unsigned)(unsigned long long)(void*)&tiles[0];
  const unsigned planeBase = (unsigned)(b * C_) * (unsigned)HW_;
  for (int i = tid; i < NTILE; i += NTHREADS) {
    int c   = i / TILE_PER_C;
    int rem = i - c * TILE_PER_C;
    int r   = rem / TCOLS;
    int col = rem - r * TCOLS;
    int gh  = h0 + r - PADK;
    int gw  = w0 + col - PADK;
    if ((unsigned)gh < (unsigned)H_ && (unsigned)gw < (unsigned)W_) {
      unsigned goff  = (planeBase + (unsigned)c * (unsigned)HW_ +
                        (unsigned)(gh * W_ + gw)) * 4u;
      unsigned laddr = ldsbase + (unsigned)i * 4u;
      asm volatile("global_load_async_to_lds_b32 %0, %1, %2"
                   :: "v"(laddr), "v"(goff), "s"(x) : "memory");
    } else {
      tiles[i] = 0.0f;   // zero padding for conv halo
    }
  }
  asm volatile("s_wait_asynccnt 0" ::: "memory");
  __syncthreads();

  // uniform coefficients
  float wkr[KS], hkr[KS];
#pragma unroll
  for (int t = 0; t < KS; t++) { wkr[t] = wk[t]; hkr[t] = hk[t]; }
  const float wbv = wbp[0], hbv = hbp[0], skv = skp[0], sbv = sbp[0];

  const int   gh   = h0 + ty, gw = w0 + tx;
  const float colf = (float)gw, rowf = (float)gh;   // rowf uniform per wave

  // ---- pass A: per-channel values (kept in registers), channel maxes,
  //      and the 6 weighted plane sums ----
  float fv[C_], wv[C_], hv[C_];
  float mf = -3.4e38f, mw = -3.4e38f, mh = -3.4e38f;
#pragma unroll
  for (int c = 0; c < C_; c++) {
    const float* Tc  = tiles + c * TILE_PER_C;
    const float  Mth = wsM[b * C_ + c] - 0.005f;
    float s = sigmoidf_(Tc[(ty + PADK) * TCOLS + (tx + PADK)]);
    float f = s * ((s > Mth ? 1.0f : 0.0f) + 0.01f);
    float wa = 0.0f, ha = 0.0f;
#pragma unroll
    for (int t = 0; t < KS; t++) {
      wa = fmaf(wkr[t], Tc[(ty + PADK) * TCOLS + (tx + t)], wa);  // XLA cross-corr along W
      ha = fmaf(hkr[t], Tc[(ty + t) * TCOLS + (tx + PADK)], ha);  // along H
    }
    float sw = fmaf(skv, sigmoidf_(wa + wbv) * (float)W_, sbv);
    float sh = fmaf(skv, sigmoidf_(ha + hbv) * (float)H_, sbv);
    fv[c] = f; wv[c] = sw; hv[c] = sh;
    mf = fmaxf(mf, f); mw = fmaxf(mw, sw); mh = fmaxf(mh, sh);
    float r0 = waveSum(f);
    float r1 = waveSum(f * f);
    float r2 = waveSum(sw * f);
    float r3 = waveSum(sh * f);
    float r4 = waveSum(colf * f);
    if (tx == 0) {                       // one lane per wave -> ds_add_f32
      atomicAdd(&accum[c*6 + 0], r0);
      atomicAdd(&accum[c*6 + 1], r1);
      atomicAdd(&accum[c*6 + 2], r2);
      atomicAdd(&accum[c*6 + 3], r3);
      atomicAdd(&accum[c*6 + 4], r4);
      atomicAdd(&accum[c*6 + 5], rowf * r0);   // row uniform in wave: factor it
    }
  }

  // ---- pass B: write channel-max-relu output straight from registers ----
  const size_t base0 = (size_t)(b * 3 * C_) * HW_ + (size_t)(gh * W_ + gw);
#pragma unroll
  for (int c = 0; c < C_; c++) {
    size_t base = base0 + (size_t)c * HW_;
    out1[base]                        = (fv[c] >= mf) ? fv[c] : 0.0f;
    out1[base + (size_t)C_   * HW_]   = (wv[c] >= mw) ? wv[c] : 0.0f;
    out1[base + (size_t)(2*C_)* HW_]  = (hv[c] >= mh) ? hv[c] : 0.0f;
  }

  __syncthreads();
  if (tid < C_*6) {
    int c = tid / 6, k = tid - c * 6;
    atomicAdd(&gacc[(b * C_ + c) * 6 + k], accum[tid]);
  }
}

// ---------------- K3: finalize bbox rows ----------------
__global__ void k_final(const float* __restrict__ gacc, float* __restrict__ res) {
  int t = threadIdx.x;
  if (t >= B_ * C_) return;
  int b = t / C_;
  const float* a = gacc + t * 6;
  float inv = 1.0f / a[0];              // 1 / sum(f)
  float sb = a[1] * inv, wb = a[2] * inv, hb = a[3] * inv;
  float cb = a[4] * inv, rb = a[5] * inv;
  float x1 = cb - 0.5f * wb, y1 = rb - 0.5f * hb;
  float x2 = cb + 0.5f * wb, y2 = rb + 0.5f * hb;
  float* r = res + t * 6;
  r[0] = (float)b; r[1] = x1; r[2] = y1; r[3] = x2; r[4] = y2; r[5] = sb;
}

extern "C" void kernel_launch(void* const* d_in, const int* in_sizes, int n_in,
                              void* d_out, int out_size, void* d_ws, size_t ws_size,
                              hipStream_t stream) {
  const float* x  = (const float*)d_in[0];
  const float* wk = (const float*)d_in[1];
  const float* wb = (const float*)d_in[2];
  const float* hk = (const float*)d_in[3];
  const float* hb = (const float*)d_in[4];
  const float* sk = (const float*)d_in[5];
  const float* sb = (const float*)d_in[6];
  float* out  = (float*)d_out;
  float* ws   = (float*)d_ws;
  float* wsM  = ws;         // 128 floats (bit-punned max)
  float* gacc = ws + 128;   // 768 floats

  k_init<<<dim3(4), dim3(256), 0, stream>>>(ws);
  k_max<<<dim3(8, 128), dim3(256), 0, stream>>>(x, (unsigned*)wsM);
  k_main<<<dim3(W_/TW, H_/TH, B_), dim3(TW, TH), 0, stream>>>(
      x, wk, wb, hk, hb, sk, sb, wsM, gacc, out);
  k_final<<<dim3(1), dim3(128), 0, stream>>>(gacc, out + OUT1_ELEMS);
}